// RoutingMaskLayer_51453708206705
// MI455X (gfx1250) — compile-verified
//
#include <hip/hip_runtime.h>

#define AS1 __attribute__((address_space(1)))
#define AS3 __attribute__((address_space(3)))

typedef float v4f __attribute__((ext_vector_type(4)));
typedef int   v4i __attribute__((vector_size(16)));   // matches builtin param type

#define N_ROUTES 4

static constexpr int kB        = 32;
static constexpr int kP        = 56 * 56;          // 3136 pixels per batch
static constexpr int kCin      = 256;              // input channels
static constexpr int kRW       = kCin / N_ROUTES;  // 64 selected channels
static constexpr int kPixF4In  = kCin / 4;         // 64 float4 per input pixel
static constexpr int kPixF4Out = kRW / 4;          // 16 float4 per output pixel
static constexpr int kN4       = kP * kPixF4Out;   // 50176 float4 per batch (output)
static constexpr int kThreads  = 256;
static constexpr int kPerThread = 4;
static constexpr int kBlocksX  = kN4 / (kThreads * kPerThread); // 49

#if __has_builtin(__builtin_amdgcn_global_load_async_to_lds_b128) && \
    __has_builtin(__builtin_amdgcn_global_store_async_from_lds_b128)
#define HAVE_ASYNC_LDS 1
#else
#define HAVE_ASYNC_LDS 0
#endif

__device__ __forceinline__ void wait_asynccnt0() {
#if __has_builtin(__builtin_amdgcn_s_wait_asynccnt)
  __builtin_amdgcn_s_wait_asynccnt(0);
#else
  asm volatile("s_wait_asynccnt 0" ::: "memory");
#endif
}

__global__ __launch_bounds__(kThreads)
void route_gather_async_kernel(const float* __restrict__ in,
                               const float* __restrict__ routing,
                               float* __restrict__ out) {
  const int b = blockIdx.y;

  // Uniform per-block argmax over 4 routes (first-max wins, like jnp.argmax).
  // Pointer and index are wave-uniform -> scalar loads + SALU.
  const float* r = routing + b * N_ROUTES;
  float best = r[0];
  int route = 0;
#pragma unroll
  for (int i = 1; i < N_ROUTES; ++i) {
    float v = r[i];
    if (v > best) { best = v; route = i; }
  }

  const int t = threadIdx.x;
  const size_t inBase  = (size_t)b * kP * kPixF4In;  // float4 units
  const size_t outBase = (size_t)b * kN4;            // float4 units
  const int idx0 = blockIdx.x * (kThreads * kPerThread);

#if HAVE_ASYNC_LDS
  // Per-lane private staging slots: 256 threads * 4 chunks * 16B = 16 KB.
  __shared__ alignas(16) char lds[kThreads * kPerThread * 16];
  const char* inB = (const char*)in;
  char* outB = (char*)out;

  size_t srcOff[kPerThread];
  size_t dstOff[kPerThread];
#pragma unroll
  for (int k = 0; k < kPerThread; ++k) {
    const int idx   = idx0 + k * kThreads + t;
    const int pixel = idx >> 4;   // /16 float4 per out pixel
    const int q     = idx & 15;
    srcOff[k] = (inBase + (size_t)pixel * kPixF4In +
                 (size_t)route * kPixF4Out + q) * 16;  // bytes
    dstOff[k] = (outBase + (size_t)idx) * 16;          // bytes
  }

  // Phase 1: HBM -> LDS, 4 async b128 loads in flight per lane (2 KB/wave).
#pragma unroll
  for (int k = 0; k < kPerThread; ++k) {
    __builtin_amdgcn_global_load_async_to_lds_b128(
        (AS1 v4i*)(inB + srcOff[k]),
        (AS3 v4i*)(lds + (size_t)(k * kThreads + t) * 16),
        0, 0);
  }
  // Only loads are outstanding -> unambiguous wait.
  wait_asynccnt0();

  // Phase 2: LDS -> HBM, fire-and-forget; s_endpgm's implicit wait-idle
  // drains ASYNCcnt before LDS deallocation.
#pragma unroll
  for (int k = 0; k < kPerThread; ++k) {
    __builtin_amdgcn_global_store_async_from_lds_b128(
        (AS1 v4i*)(outB + dstOff[k]),
        (AS3 v4i*)(lds + (size_t)(k * kThreads + t) * 16),
        0, 0);
  }
#else
  // Fallback: direct 128-bit nontemporal copy (streaming, no reuse).
  const v4f* in4 = (const v4f*)in;
  v4f* out4 = (v4f*)out;
  v4f vals[kPerThread];
#pragma unroll
  for (int k = 0; k < kPerThread; ++k) {
    const int idx   = idx0 + k * kThreads + t;
    const int pixel = idx >> 4;
    const int q     = idx & 15;
    vals[k] = __builtin_nontemporal_load(
        in4 + inBase + (size_t)pixel * kPixF4In + (size_t)route * kPixF4Out + q);
  }
#pragma unroll
  for (int k = 0; k < kPerThread; ++k) {
    const int idx = idx0 + k * kThreads + t;
    __builtin_nontemporal_store(vals[k], out4 + outBase + idx);
  }
#endif
}

extern "C" void kernel_launch(void* const* d_in, const int* in_sizes, int n_in,
                              void* d_out, int out_size, void* d_ws, size_t ws_size,
                              hipStream_t stream) {
  const float* in      = (const float*)d_in[0];   // [32,56,56,256] f32
  const float* routing = (const float*)d_in[1];   // [32,4] f32
  float* out           = (float*)d_out;           // [32,56,56,64] f32

  dim3 grid(kBlocksX, kB, 1);
  route_gather_async_kernel<<<grid, dim3(kThreads, 1, 1), 0, stream>>>(
      in, routing, out);
}